// ModelNew_66657892434264
// MI455X (gfx1250) — compile-verified
//
#include <hip/hip_runtime.h>

typedef __attribute__((ext_vector_type(2))) float v2f;
typedef __attribute__((ext_vector_type(8))) float v8f;
typedef int v4i_vs __attribute__((__vector_size__(4 * sizeof(int))));  // b128 builtin param

#define NN 4096
#define BM 128
#define BN 128
#define BK 32
#define LDA_P 36   // A pad: conflict-free 16-lane stride reads, 16B-aligned rows
#define PB    288  // B swizzle block pitch (dwords): +288 = +32 banks -> halves disjoint

#define AS3 __attribute__((address_space(3)))

#if defined(__AMDGCN__) && \
    __has_builtin(__builtin_amdgcn_global_load_async_to_lds_b128) && \
    __has_builtin(__builtin_amdgcn_global_load_async_to_lds_b32)
#define HAVE_ASYNC_LDS 1
#else
#define HAVE_ASYNC_LDS 0
#endif

__device__ __forceinline__ void wait_async_zero() {
#if HAVE_ASYNC_LDS
#if __has_builtin(__builtin_amdgcn_s_wait_asynccnt)
    __builtin_amdgcn_s_wait_asynccnt(0);
#else
    asm volatile("s_wait_asynccnt 0x0" ::: "memory");
#endif
#endif
}

__global__ __launch_bounds__(256)
void triu_sgemm_wmma_f32(const float* __restrict__ A,
                         const float* __restrict__ B,
                         float* __restrict__ C) {
    const int J = blockIdx.x;   // column tile
    const int I = blockIdx.y;   // row tile
    const int tid = threadIdx.x;
    const int rowBase = I * BM;
    const int colBase = J * BN;

    if (I > J) {
        // Strict lower-triangular tile of C is zero; d_out is poisoned so write it.
        const float4 z = make_float4(0.f, 0.f, 0.f, 0.f);
        #pragma unroll
        for (int i = 0; i < 16; ++i) {
            const int idx = (i * 256 + tid) * 4;
            const int r = idx >> 7;
            const int c = idx & 127;
            *(float4*)(C + (size_t)(rowBase + r) * NN + colBase + c) = z;
        }
        return;
    }

    __shared__ float sA[2][BM * LDA_P];      // row-major, padded
    __shared__ float sBw[2][(BK / 2) * PB];  // pair-swizzled: (k,n) -> (k>>1)*PB + 2n + (k&1)

    const int wave  = tid >> 5;
    const int lane  = tid & 31;
    const int lhalf = lane >> 4;
    const int l16   = lane & 15;

    // A staging map: 2 threads/row, 2x b128-pairs each
    const int a_r = tid >> 1;         // 0..127
    const int a_c = (tid & 1) * 16;   // 0 or 16
    // B staging map: wave w covers rows 4w..4w+3; instr (ri,cg): row 4w+ri, cols cg*32+lane
    const int b_row0 = wave * 4;

    // Triangular k-range: A[i][k]==0 for k<i, B[k][j]==0 for k>j.
    const int kBeg = rowBase;
    const int kEnd = colBase + BN;

    auto stage = [&](int buf, int kt) {
        const float* gA = A + (size_t)(rowBase + a_r) * NN + kt + a_c;
        float*       lA = &sA[buf][a_r * LDA_P + a_c];
#if HAVE_ASYNC_LDS
        #pragma unroll
        for (int v = 0; v < 4; ++v)
            __builtin_amdgcn_global_load_async_to_lds_b128(
                (v4i_vs*)(gA + v * 4), (AS3 v4i_vs*)(lA + v * 4), 0, 0);
        #pragma unroll
        for (int m = 0; m < 16; ++m) {
            const int ri = m >> 2, cg = m & 3;
            const int rr = b_row0 + ri;                 // 0..31
            const int c  = cg * 32 + lane;              // 0..127
            const float* gB = B + (size_t)(kt + rr) * NN + colBase + c;
            float*       lB = &sBw[buf][(rr >> 1) * PB + c * 2 + (rr & 1)];
            __builtin_amdgcn_global_load_async_to_lds_b32(
                (int*)gB, (AS3 int*)lB, 0, 0);
        }
#else
        #pragma unroll
        for (int v = 0; v < 4; ++v)
            *(float4*)(lA + v * 4) = *(const float4*)(gA + v * 4);
        #pragma unroll
        for (int m = 0; m < 16; ++m) {
            const int ri = m >> 2, cg = m & 3;
            const int rr = b_row0 + ri;
            const int c  = cg * 32 + lane;
            sBw[buf][(rr >> 1) * PB + c * 2 + (rr & 1)] =
                B[(size_t)(kt + rr) * NN + colBase + c];
        }
        __builtin_prefetch(gA + BK, 0, 1);              // global_prefetch_b8
#endif
    };

    v8f acc[8];
    const v8f vzero = {0.f, 0.f, 0.f, 0.f, 0.f, 0.f, 0.f, 0.f};
    #pragma unroll
    for (int j = 0; j < 8; ++j) acc[j] = vzero;

    int cur = 0;
    stage(0, kBeg);                       // prologue: tile 0 -> buf 0

    for (int kt = kBeg; kt < kEnd; kt += BK) {
        wait_async_zero();                // my async writes to buf[cur] landed
        __syncthreads();                  // all tile data visible; buf[cur^1] free
        if (kt + BK < kEnd)
            stage(cur ^ 1, kt + BK);      // DMA next tile while we compute

        const int am = (wave * 16 + l16) * LDA_P;
        const int bn = (l16 << 1);        // 2n within swizzle block
        #pragma unroll
        for (int kk = 0; kk < BK; kk += 4) {
            v2f a;                        // A 16x4 frag: lanes0-15 K={kk,kk+1}, lanes16-31 K={kk+2,kk+3}
            a.x = sA[cur][am + kk + 2 * lhalf + 0];
            a.y = sA[cur][am + kk + 2 * lhalf + 1];
            const int bblk = (kk / 2 + lhalf) * PB;     // rows {kk+2h, kk+2h+1}
            #pragma unroll
            for (int j = 0; j < 8; ++j) {
                // single aligned ds_load_b64 -> {B[k0][n], B[k0+1][n]}
                v2f b = *(const v2f*)&sBw[cur][bblk + j * 32 + bn];
                acc[j] = __builtin_amdgcn_wmma_f32_16x16x4_f32(
                    false, a, false, b, (short)0, acc[j], false, false);
            }
        }
        cur ^= 1;
    }

    // Store 16x128 strip; mask below-diagonal on diagonal tiles.
    #pragma unroll
    for (int j = 0; j < 8; ++j) {
        const int col = colBase + j * 16 + l16;
        #pragma unroll
        for (int r = 0; r < 8; ++r) {
            const int row = rowBase + wave * 16 + r + 8 * lhalf;
            float val = acc[j][r];
            if (I == J && row > col) val = 0.f;
            C[(size_t)row * NN + col] = val;
        }
    }
}

extern "C" void kernel_launch(void* const* d_in, const int* in_sizes, int n_in,
                              void* d_out, int out_size, void* d_ws, size_t ws_size,
                              hipStream_t stream) {
    (void)in_sizes; (void)n_in; (void)out_size; (void)d_ws; (void)ws_size;
    const float* A = (const float*)d_in[0];
    const float* B = (const float*)d_in[1];
    float*       C = (float*)d_out;
    dim3 grid(NN / BN, NN / BM);   // 32 x 32 tiles
    triu_sgemm_wmma_f32<<<grid, dim3(256), 0, stream>>>(A, B, C);
}